// TopologicalLoss_65455301591738
// MI455X (gfx1250) — compile-verified
//
#include <hip/hip_runtime.h>
#include <stdint.h>

// Problem shape (fixed by the reference): B=8, L=1024, C=8192
#define BB 8
#define LL 1024
#define CC 8192
#define NROWS (BB * LL)          // 8192 rows of grad
#define NGRAD (NROWS * CC)       // 67,108,864 grad elements
#define FD_EPS 1e-3f
#define FILL_THREADS 1024        // 32 waves: stage 32 KB with 2 b128/thread

typedef unsigned int uv4 __attribute__((ext_vector_type(4)));
typedef int          iv4 __attribute__((ext_vector_type(4)));
typedef int          iv8 __attribute__((ext_vector_type(8)));

#if defined(__has_builtin)
#  if __has_builtin(__builtin_amdgcn_tensor_store_from_lds)
#    define HAVE_TDM 1
#  endif
#endif

#if defined(__clang_major__) && (__clang_major__ >= 23)
#  define TDM_STORE(g0, g1, g2, g3, cpol)                                      \
     do { iv8 _gz = (iv8){0,0,0,0,0,0,0,0};                                    \
          __builtin_amdgcn_tensor_store_from_lds((g0),(g1),(g2),(g3),_gz,(cpol)); } while (0)
#else
#  define TDM_STORE(g0, g1, g2, g3, cpol)                                      \
     __builtin_amdgcn_tensor_store_from_lds((g0),(g1),(g2),(g3),(cpol))
#endif

// ---------------------------------------------------------------------------
// Broadcast fill of grad via the Tensor Data Mover.
//
// Verified lowering (round 4 probe):
//   tensor_store_from_lds s[g0:g0+3], s[g1:g1+7] th:TH_STORE_NT
//   s_wait_tensorcnt 0x0
//
// One block per (b,l) row. The whole row is one scalar v broadcast over
// C=8192 floats (32 KB). Stage v into LDS (2x ds_store_b128 per thread at
// 1024 threads), then ONE tensor_store_from_lds DMAs the 32 KB tile to
// global memory. TDM takes a byte-granular global address, so the +4B
// offset of grad inside d_out (loss occupies d_out[0]) costs nothing.
// cpol=1 -> TH_STORE_NT: the 256 MB output exceeds the 192 MB L2 and is
// never re-read, so stream it to HBM without rinsing L2.
// ---------------------------------------------------------------------------
__global__ void __launch_bounds__(FILL_THREADS)
topo_fill_kernel(const int* __restrict__ mask,
                 const float* __restrict__ diff,
                 float* __restrict__ gradbase /* d_out + 1 */) {
    __shared__ float tile[CC];           // 32 KB of the 320 KB WGP LDS

    const int row = blockIdx.x;          // flat (b*L + l)
    const int b   = row >> 10;           // L = 1024
    const float dv = diff[b];            // scalar (s_load) — row is uniform
    const float v  = mask[row] ? dv : 0.0f;

    // Stage the broadcast value: 1024 threads x 2 float4 = 8192 floats.
    float4  v4 = make_float4(v, v, v, v);
    float4* t4 = (float4*)tile;
    t4[threadIdx.x]                = v4;   // ds_store_b128
    t4[threadIdx.x + FILL_THREADS] = v4;   // ds_store_b128 offset:16384
    __syncthreads();

#if defined(HAVE_TDM)
    if (threadIdx.x == 0) {
        const uint64_t gaddr = (uint64_t)(uintptr_t)(gradbase + (size_t)row * CC);
        const uint32_t laddr = (uint32_t)(uintptr_t)&tile[0];  // low 32b = LDS offset

        // ---- D# group 0 (128b): count=1, lds_addr, global_addr[56:0], type=2
        uv4 g0;
        g0[0] = 1u;                                        // count=1, user store D#
        g0[1] = laddr;                                     // lds_addr
        g0[2] = (uint32_t)(gaddr & 0xFFFFFFFFu);           // global_addr[31:0]
        g0[3] = (uint32_t)((gaddr >> 32) & 0x01FFFFFFu)    // global_addr[56:32]
              | (2u << 30);                                // type = 2 ("image")

        // ---- D# group 1 (256b): data_size=4B, 2D tensor 8192x1, tile 8192x1
        iv8 g1;
        g1[0] = (2 << 16);                  // data_size = 4 bytes; mask/flags = 0
        g1[1] = (int)((CC & 0xFFFF) << 16); // tensor_dim0[15:0] (8192)
        g1[2] = (int)(((CC >> 16) & 0xFFFF) | (1 << 16)); // dim0 hi | tensor_dim1=1
        g1[3] = (int)((CC & 0xFFFF) << 16); // tensor_dim1 hi=0 | tile_dim0 = 8192
        g1[4] = 1;                          // tile_dim1 = 1, tile_dim2 = 0 (unused)
        g1[5] = CC;                         // tensor_dim0_stride[31:0] = 8192
        g1[6] = 0;                          // stride hi | tensor_dim1_stride lo
        g1[7] = 0;

        iv4 g2 = (iv4){0, 0, 0, 0};        // dims 2..4 unused (2D tile)
        iv4 g3 = (iv4){0, 0, 0, 0};

        TDM_STORE(g0, g1, g2, g3, /*cpol -> TH_STORE_NT*/ 1);
    }
    // Drain the tensor DMA before the wave (and its LDS tile) retires.
    __builtin_amdgcn_s_wait_tensorcnt(0);
#else
    // Fallback path (no TDM builtin): plain scalar stores (grad base is only
    // 4-byte aligned inside d_out, so no b128 vector stores here).
    float* gr = gradbase + (size_t)row * CC;
    for (int k = threadIdx.x; k < CC; k += blockDim.x)
        gr[k] = tile[k];
#endif
}

// ---------------------------------------------------------------------------
// Tiny scalar prologue:
//   loss        -> d_out[0]
//   diff[0..7]  -> d_ws (floats)
// 16 input floats total; one lane does it serially.
// ---------------------------------------------------------------------------
__global__ void topo_prep_kernel(const float* __restrict__ target,
                                 float* __restrict__ out,
                                 float* __restrict__ diff) {
    if (threadIdx.x == 0) {
        float per[BB];
        float s = 0.0f;
        #pragma unroll
        for (int b = 0; b < BB; ++b) {
            float t0 = target[2 * b + 0];
            float t1 = target[2 * b + 1];
            float e0 = (1.0f - t0) * (1.0f - t0);
            float e1 = t1 * t1;
            per[b] = 0.5f * (e0 + e1);
            s += e0 + e1;
        }
        float loss = s * (1.0f / (2.0f * BB));
        out[0] = loss;
        #pragma unroll
        for (int b = 0; b < BB; ++b)
            diff[b] = (per[b] - loss) * (1.0f / FD_EPS);
    }
}

// ---------------------------------------------------------------------------
// Launch: prep (serial, tiny) then the TDM broadcast fill.
// d_in[0] = logits (unused by the math), d_in[1] = mask, d_in[2] = target.
// d_out = [loss (1 float)] ++ [grad (B*L*C floats)].
// ---------------------------------------------------------------------------
extern "C" void kernel_launch(void* const* d_in, const int* in_sizes, int n_in,
                              void* d_out, int out_size, void* d_ws, size_t ws_size,
                              hipStream_t stream) {
    (void)in_sizes; (void)n_in; (void)out_size; (void)ws_size;
    const int*   mask   = (const int*)d_in[1];
    const float* target = (const float*)d_in[2];
    float*       out    = (float*)d_out;
    float*       diff   = (float*)d_ws;   // 8 floats of scratch

    topo_prep_kernel<<<1, 32, 0, stream>>>(target, out, diff);
    topo_fill_kernel<<<NROWS, FILL_THREADS, 0, stream>>>(mask, diff, out + 1);
}